// loss_26405458936156
// MI455X (gfx1250) — compile-verified
//
#include <hip/hip_runtime.h>
#include <hip/hip_bf16.h>

typedef __attribute__((ext_vector_type(2))) float v2f;
typedef __attribute__((ext_vector_type(8))) float v8f;

#define TILE_LD 33   // 32 columns + 1 pad to dodge bank conflicts

// Column-sum of a 32-row x 16-col slab of an LDS tile via fp32 WMMA:
// A = ones(16x4)  =>  D[i][j] (+)= sum_k B[k][j].  8 chained WMMAs cover rows 0..31.
// B layout (16x4 dual of the documented A 16x4 layout): lane = n + 16*h holds
// elements (k = v + 2h, n) in VGPR v. Any k-permutation within a column is
// sum-invariant, so only col = lane&15 must be right.
__device__ __forceinline__ v8f colsum_accum(v8f acc, const float* tile, int colBase, int lane) {
  v2f a; a[0] = 1.0f; a[1] = 1.0f;
  const int h = lane >> 4;
  const int n = lane & 15;
#pragma unroll
  for (int kc = 0; kc < 8; ++kc) {
    const int r0 = 4 * kc + 2 * h;
    v2f bm;
    bm[0] = tile[r0 * TILE_LD + colBase + n];
    bm[1] = tile[(r0 + 1) * TILE_LD + colBase + n];
    acc = __builtin_amdgcn_wmma_f32_16x16x4_f32(
        /*neg_a=*/false, a, /*neg_b=*/false, bm,
        /*c_mod=*/(short)0, acc, /*reuse_a=*/false, /*reuse_b=*/false);
  }
  return acc;
}

__global__ __launch_bounds__(256)
void loss_26405458936156_kernel(const float* __restrict__ pred,
                                const int* __restrict__ truth,
                                const int* __restrict__ s_ptr,
                                float* __restrict__ out,
                                int size0, int size1) {
  __shared__ float Ld[32 * TILE_LD];  // dis_loss values, zero-padded to 32x32
  __shared__ float Lp[32 * TILE_LD];  // p_loss values,   zero-padded to 32x32

  const int S = s_ptr[0];             // low word correct for int32 or LE int64
  const int C = S * S;
  const int B = size0 / (3 * C);
  const int N = size1 / (2 * B);
  if (C > 32 || B > 32 || N > 32) return;  // uniform early-out (unsupported cfg)

  const int tid    = threadIdx.x;
  const int wave   = tid >> 5;
  const int lane   = tid & 31;
  const int nwaves = blockDim.x >> 5;

  for (int i = tid; i < 32 * TILE_LD; i += blockDim.x) { Ld[i] = 0.0f; Lp[i] = 0.0f; }
  __syncthreads();

  const float cell  = 80.0f / (float)S;
  const float scale = (float)S / 80.0f;

  // Phase 1: one wave per batch row b; lane plays both roles (truth idx n for
  // lane<N, cell idx c for lane<C). Cross-lane traffic via wave32 shuffles.
  for (int b = wave; b < B; b += nwaves) {
    float tx = 0.0f, ty = 0.0f;
    int lab = 0;
    if (lane < N) {
      const float t0 = (float)truth[(b * N + lane) * 2 + 0];
      const float t1 = (float)truth[(b * N + lane) * 2 + 1];
      tx  = fmodf(t0 + 14.0f, cell) * scale;
      ty  = fmodf(t1 + 14.0f, cell) * scale;
      lab = (int)floorf((t0 + 14.0f) / cell) * S + (int)floorf((t1 + 14.0f) / cell);
    }
    const float txl = __shfl(tx, N - 1);   // tx_last / ty_last (broadcast lane N-1)
    const float tyl = __shfl(ty, N - 1);

    // counts[c]==N  <=>  every label equals c  (full wave executes the shuffles)
    bool match = true;
    for (int n = 0; n < N; ++n) {
      match = match && (__shfl(lab, n) == lane);
    }

    const int c = lane;
    float pv = 0.0f, dv = 0.0f;
    if (c < C) {
      const float conf = pred[b * 3 * C + 3 * c + 0];
      const float cx   = pred[b * 3 * C + 3 * c + 1];
      const float cy   = pred[b * 3 * C + 3 * c + 2];
      const float dx = fabsf(cx * 16.0f - txl * 16.0f);
      const float dy = fabsf(cy * 16.0f - tyl * 16.0f);
      const float iou = fmaxf(28.0f - 2.0f * dx, 0.0f) * fmaxf(28.0f - 2.0f * dy, 0.0f)
                        / ((28.0f + dx) * (28.0f + dy));
      const float cd = conf - iou;
      pv = match ? cd * cd : 0.5f * conf * conf;
      if (c < N) {  // lane c == truth index n here, so lab/tx/ty are its own
        const float px = pred[b * 3 * C + 3 * lab + 1];
        const float py = pred[b * 3 * C + 3 * lab + 2];
        dv = 5.0f * ((px - tx) * (px - tx) + (py - ty) * (py - ty));
      }
    }
    Lp[b * TILE_LD + c] = pv;
    Ld[b * TILE_LD + c] = dv;
  }
  __syncthreads();

  // Phase 2: wave 0 only (EXEC all-ones => WMMA-legal). fp32 WMMA reductions.
  if (wave == 0) {
    v8f accP = {};                              // both column groups -> total sum
    accP = colsum_accum(accP, Lp, 0, lane);
    accP = colsum_accum(accP, Lp, 16, lane);
    v8f accD0 = {};                             // dis colsums, cols 0..15
    accD0 = colsum_accum(accD0, Ld, 0, lane);
    v8f accD1 = {};                             // dis colsums, cols 16..31
    accD1 = colsum_accum(accD1, Ld, 16, lane);

    // accP[0]: lanes 0-15 = row0, lanes 16-31 = row8 (identical since A=ones),
    // so an xor-reduction within each 16-lane half yields the total in every lane.
    float tP = accP[0];
    tP += __shfl_xor(tP, 8);
    tP += __shfl_xor(tP, 4);
    tP += __shfl_xor(tP, 2);
    tP += __shfl_xor(tP, 1);

    // Lane l: colsum of dis column l (rows of D are identical, so lanes 16-31
    // read their M=8 copy of cols 16..31 directly).
    const float cs = (lane < 16) ? accD0[0] : accD1[0];
    if (lane < C) {
      out[lane] = cs / (float)C + tP / ((float)B * (float)C);
    }
  }
}

extern "C" void kernel_launch(void* const* d_in, const int* in_sizes, int n_in,
                              void* d_out, int out_size, void* d_ws, size_t ws_size,
                              hipStream_t stream) {
  (void)n_in; (void)out_size; (void)d_ws; (void)ws_size;
  const float* pred  = (const float*)d_in[0];
  const int*   truth = (const int*)d_in[1];
  const int*   s_ptr = (const int*)d_in[2];
  float*       out   = (float*)d_out;
  // Tiny, latency-bound problem: one kernel, one workgroup (8 wave32s) does
  // everything; phase-2 reduction runs on the WMMA pipe in full fp32.
  loss_26405458936156_kernel<<<1, 256, 0, stream>>>(pred, truth, s_ptr, out,
                                                    in_sizes[0], in_sizes[1]);
}